// NumAttention_16329465660072
// MI455X (gfx1250) — compile-verified
//
#include <hip/hip_runtime.h>

// Problem constants (reference: B,P,DC,DN,H,DH = 4,2048,512,64,8,64)
#define Bn   4
#define Pn   2048
#define DCn  512
#define DNn  64
#define Hn   8
#define DHn  64
#define NVn  8            // DC / DH
#define CHn  64           // chunk (rows per block)
#define NCH  (Pn / CHn)   // 32 chunks
#define PIT  65           // LDS pitch (odd -> conflict-free column walks)

typedef __attribute__((ext_vector_type(2))) float v2f;
typedef __attribute__((ext_vector_type(8))) float v8f;

__device__ __forceinline__ void softmax_serial(const float* __restrict__ src,
                                               float* __restrict__ dst, int n) {
  float m = src[0];
  for (int j = 1; j < n; ++j) m = fmaxf(m, src[j]);
  float s = 0.f;
  for (int j = 0; j < n; ++j) { float e = __expf(src[j] - m); dst[j] = e; s += e; }
  float r = 1.0f / s;
  for (int j = 0; j < n; ++j) dst[j] *= r;
}

// ---------------------------------------------------------------------------
// K1: per (b, head, chunk) compute chunk state G[h] = sum_n v[n] * k[n,h]
// ---------------------------------------------------------------------------
__global__ __launch_bounds__(256) void numattn_state_k1(
    const float* __restrict__ x_cat, const float* __restrict__ x_num,
    const float* __restrict__ W_K, const float* __restrict__ W_V,
    float* __restrict__ G) {
  __shared__ float pk_s[NVn];
  __shared__ float pv_s[DNn];
  __shared__ float kS[CHn * PIT];
  __shared__ float vS[CHn];

  const int tid = threadIdx.x;
  const int c   = blockIdx.x % NCH;
  const int bi  = blockIdx.x / NCH;
  const int i   = bi % Hn;
  const int b   = bi / Hn;
  const int p0  = c * CHn;

  if (tid == 0)  softmax_serial(W_K + i * NVn, pk_s, NVn);
  if (tid == 32) softmax_serial(W_V + i * DNn, pv_s, DNn);
  __syncthreads();

  // k[r,h] = sum_v pk[v] * x_cat[b, p0+r, v*64 + h]   (coalesced: h fast)
  #pragma unroll
  for (int it = 0; it < (CHn * DHn) / 256; ++it) {
    int idx = it * 256 + tid;
    int r = idx >> 6, h = idx & 63;
    const float* xp = x_cat + ((size_t)(b * Pn + p0 + r) * DCn) + h;
    float ak = 0.f;
    #pragma unroll
    for (int v = 0; v < NVn; ++v) ak += pk_s[v] * xp[v * DHn];
    kS[r * PIT + h] = ak;
  }
  if (tid < CHn) {
    const float* xn = x_num + (size_t)(b * Pn + p0 + tid) * DNn;
    float acc = 0.f;
    #pragma unroll 8
    for (int d = 0; d < DNn; ++d) acc += pv_s[d] * xn[d];
    vS[tid] = acc;
  }
  __syncthreads();

  if (tid < DHn) {
    float g = 0.f;
    #pragma unroll 8
    for (int n = 0; n < CHn; ++n) g += vS[n] * kS[n * PIT + tid];
    G[(bi * NCH + c) * DHn + tid] = g;
  }
}

// ---------------------------------------------------------------------------
// K2: per (b, head, chunk): Sinit = prefix of G; q = xq@pp (WMMA);
//     scores = q@k^T (WMMA); z = q.Sinit + tril(scores).v
// ---------------------------------------------------------------------------
__global__ __launch_bounds__(256) void numattn_main_k2(
    const float* __restrict__ x_cat, const float* __restrict__ x_num,
    const float* __restrict__ W_K, const float* __restrict__ W_Q,
    const float* __restrict__ W_pred, const float* __restrict__ W_V,
    const float* __restrict__ G, float* __restrict__ out) {
  extern __shared__ float smem[];
  float* ppS = smem;                  // DHn * PIT  softmax(W_pred[i])
  float* kS  = ppS + DHn * PIT;       // CHn * PIT
  float* xqS = kS  + CHn * PIT;       // CHn * PIT  (scores overlay after Phase C)
  float* qS  = xqS + CHn * PIT;       // CHn * PIT
  __shared__ float pk_s[NVn], pq_s[NVn], pv_s[DNn];
  __shared__ float vS[CHn], Sini[DHn];

  const int tid = threadIdx.x;
  const int c   = blockIdx.x % NCH;
  const int bi  = blockIdx.x / NCH;
  const int i   = bi % Hn;
  const int b   = bi / Hn;
  const int p0  = c * CHn;

  if (tid == 0)  softmax_serial(W_K + i * NVn, pk_s, NVn);
  if (tid == 32) softmax_serial(W_Q + i * NVn, pq_s, NVn);
  if (tid == 64) softmax_serial(W_V + i * DNn, pv_s, DNn);
  if (tid < DHn)
    softmax_serial(W_pred + (size_t)(i * DHn + tid) * DHn, ppS + tid * PIT, DHn);
  if (tid < DHn) {                    // cross-chunk prefix state
    float s = 0.f;
    for (int sg = 0; sg < c; ++sg) s += G[(bi * NCH + sg) * DHn + tid];
    Sini[tid] = s;
  }
  __syncthreads();

  // ---- Phase A: k, xq (block-mix of x_cat), v ----
  #pragma unroll
  for (int it = 0; it < (CHn * DHn) / 256; ++it) {
    int idx = it * 256 + tid;
    int r = idx >> 6, h = idx & 63;
    const float* xp = x_cat + ((size_t)(b * Pn + p0 + r) * DCn) + h;
    float ak = 0.f, aq = 0.f;
    #pragma unroll
    for (int v = 0; v < NVn; ++v) {
      float xv = xp[v * DHn];
      ak += pk_s[v] * xv;
      aq += pq_s[v] * xv;
    }
    kS[r * PIT + h]  = ak;
    xqS[r * PIT + h] = aq;
  }
  if (tid < CHn) {
    const float* xn = x_num + (size_t)(b * Pn + p0 + tid) * DNn;
    float acc = 0.f;
    #pragma unroll 8
    for (int d = 0; d < DNn; ++d) acc += pv_s[d] * xn[d];
    vS[tid] = acc;
  }
  __syncthreads();

  const int wave = tid >> 5, lane = tid & 31;
  const int hs = lane >> 4, l = lane & 15;   // half-wave select / lane-in-half

  // ---- Phase B: q(64x64) = xq(64x64) @ pp(64x64), fp32 WMMA 16x16x4 ----
  #pragma unroll
  for (int rep = 0; rep < 2; ++rep) {
    int tile = wave * 2 + rep;               // 16 tiles, 2 per wave
    int m0 = (tile >> 2) * 16, n0 = (tile & 3) * 16;
    v8f acc = {};
    #pragma unroll
    for (int kk = 0; kk < DHn; kk += 4) {
      int ka = kk + 2 * hs;                  // lanes0-15: K=kk,kk+1; 16-31: kk+2,kk+3
      v2f a, bb;
      a.x  = xqS[(m0 + l) * PIT + ka];
      a.y  = xqS[(m0 + l) * PIT + ka + 1];
      bb.x = ppS[ka * PIT + n0 + l];
      bb.y = ppS[(ka + 1) * PIT + n0 + l];
      acc = __builtin_amdgcn_wmma_f32_16x16x4_f32(false, a, false, bb,
                                                  (short)0, acc, false, false);
    }
    #pragma unroll
    for (int j = 0; j < 8; ++j)
      qS[(m0 + hs * 8 + j) * PIT + n0 + l] = acc[j];   // D layout: VGPR j -> M=j / j+8
  }
  __syncthreads();

  // ---- Phase C: scores(64x64) = q @ k^T, fp32 WMMA; overlay into xqS ----
  #pragma unroll
  for (int rep = 0; rep < 2; ++rep) {
    int tile = wave * 2 + rep;
    int m0 = (tile >> 2) * 16, n0 = (tile & 3) * 16;
    v8f acc = {};
    #pragma unroll
    for (int kk = 0; kk < DHn; kk += 4) {
      int ka = kk + 2 * hs;
      v2f a, bb;
      a.x  = qS[(m0 + l) * PIT + ka];
      a.y  = qS[(m0 + l) * PIT + ka + 1];
      bb.x = kS[(n0 + l) * PIT + ka];        // B[k][n] = k-matrix[n][k]
      bb.y = kS[(n0 + l) * PIT + ka + 1];
      acc = __builtin_amdgcn_wmma_f32_16x16x4_f32(false, a, false, bb,
                                                  (short)0, acc, false, false);
    }
    #pragma unroll
    for (int j = 0; j < 8; ++j)
      xqS[(m0 + hs * 8 + j) * PIT + n0 + l] = acc[j];
  }
  __syncthreads();

  // ---- Phase D: z[m] = q[m]·Sinit + sum_{n<=m} scores[m][n]*v[n] ----
  if (tid < CHn) {
    int m = tid;
    float z = 0.f;
    #pragma unroll 8
    for (int h = 0; h < DHn; ++h) z += qS[m * PIT + h] * Sini[h];
    for (int n = 0; n <= m; ++n)  z += xqS[m * PIT + n] * vS[n];
    out[(size_t)(b * Pn + p0 + m) * Hn + i] = z;   // layout [B, Q, H]
  }
}

// ---------------------------------------------------------------------------
extern "C" void kernel_launch(void* const* d_in, const int* in_sizes, int n_in,
                              void* d_out, int out_size, void* d_ws, size_t ws_size,
                              hipStream_t stream) {
  const float* x_cat  = (const float*)d_in[0];
  const float* x_num  = (const float*)d_in[1];
  const float* W_K    = (const float*)d_in[2];
  const float* W_Q    = (const float*)d_in[3];
  const float* W_pred = (const float*)d_in[4];
  const float* W_V    = (const float*)d_in[5];
  float* out = (float*)d_out;
  float* G   = (float*)d_ws;   // Bn*Hn*NCH*DHn floats = 256 KB

  dim3 grid(Bn * Hn * NCH);    // 1024 blocks
  dim3 block(256);             // 8 wave32 waves

  numattn_state_k1<<<grid, block, 0, stream>>>(x_cat, x_num, W_K, W_V, G);

  size_t smem = (size_t)(DHn * PIT + 3 * CHn * PIT) * sizeof(float);  // ~66.5 KB dynamic LDS
  numattn_main_k2<<<grid, block, smem, stream>>>(x_cat, x_num, W_K, W_Q,
                                                 W_pred, W_V, G, out);
}